// RGCNHeteroGNN_5686536699973
// MI455X (gfx1250) — compile-verified
//
#include <hip/hip_runtime.h>
#include <hip/hip_bf16.h>

typedef __attribute__((ext_vector_type(2))) float v2f;
typedef __attribute__((ext_vector_type(8))) float v8f;

#define N_ADDR 400000
#define N_TX   400000
#define NEDGE  1500000
#define F_IN   64
#define HID    32

// ---------------------------------------------------------------------------
// GEMM: Y[N,32] = X[N,K] @ W[K,32] using V_WMMA_F32_16X16X4_F32 (f32-exact).
// One wave computes a 16x32 strip (two 16x16 WMMA accumulators).
// Block = 256 threads = 8 waves = 128 rows. N must be a multiple of 128
// (400000 = 3125*128, exact), so EXEC is all-ones for every WMMA.
//
// W is staged into LDS **pre-swizzled into B-fragment order**: for k-step ks,
// col-tile t, lane L the fragment pair {W[kk][col], W[kk+1][col]} (kk =
// 4*ks + (L>>4)*2, col = t*16 + (L&15)) is stored contiguously as a float2 at
// sB[(ks*2+t)*32 + L]. The inner loop then needs exactly one conflict-free
// ds_load_b64 per tile with no register repacking before the WMMA.
// ---------------------------------------------------------------------------
__global__ __launch_bounds__(256) void gemm_x32_wmma(
    const float* __restrict__ X, const float* __restrict__ W,
    float* __restrict__ Y, int N, int K)
{
    __shared__ v2f sB[(F_IN / 4) * 2 * 32];   // max 1024 x 8B = 8KB
    int tid = threadIdx.x;

    // Stage W in fragment order: entries = (K/4)*2*32 = K*16
    for (int idx = tid; idx < K * 16; idx += 256) {
        int ks   = idx >> 6;            // k-step
        int t    = (idx >> 5) & 1;      // col tile
        int ln   = idx & 31;            // lane it will feed
        int kk   = 4 * ks + ((ln >> 4) << 1);
        int col  = t * 16 + (ln & 15);
        v2f w;
        w.x = W[kk * HID + col];
        w.y = W[(kk + 1) * HID + col];
        sB[idx] = w;
    }
    __syncthreads();

    int wave = tid >> 5;
    int lane = tid & 31;
    int rowBase = blockIdx.x * 128 + wave * 16;
    int m       = lane & 15;          // row within tile (A) / col within tile (B,D)
    int khalf   = (lane >> 4) * 2;    // lanes 16-31 hold K+2,K+3 halves

    v8f acc0 = {};
    v8f acc1 = {};
    const float* Xrow = X + (size_t)(rowBase + m) * K;
    const v2f*   bRow = sB + lane;

    for (int ks = 0; ks < (K >> 2); ++ks) {
        int kk = (ks << 2) + khalf;
        // A fragment: 16x4 f32 — row m, cols kk..kk+1 (8B aligned)
        v2f a  = *(const v2f*)(Xrow + kk);
        // B fragments: single ds_load_b64 per tile, pre-swizzled
        v2f b0 = bRow[(ks * 2 + 0) * 32];
        v2f b1 = bRow[(ks * 2 + 1) * 32];
        acc0 = __builtin_amdgcn_wmma_f32_16x16x4_f32(false, a, false, b0,
                                                     (short)0, acc0, false, false);
        acc1 = __builtin_amdgcn_wmma_f32_16x16x4_f32(false, a, false, b1,
                                                     (short)0, acc1, false, false);
    }

    // D layout: VGPR r -> M=r (lanes 0-15) / M=r+8 (lanes 16-31), N = lane&15
    int mOff = (lane < 16) ? 0 : 8;
    for (int r = 0; r < 8; ++r) {
        size_t row = (size_t)(rowBase + mOff + r);
        Y[row * HID + m]      = acc0[r];
        Y[row * HID + 16 + m] = acc1[r];
    }
}

// ---------------------------------------------------------------------------
// Edge pass: one wave per edge, 32 lanes = 32 features. Coalesced gather of
// projected src features + coalesced global_atomic_add_f32 scatter into acc.
// Node tables are 51.2MB -> L2-resident on the 192MB L2.
// ---------------------------------------------------------------------------
__global__ __launch_bounds__(256) void scatter_kernel(
    const int* __restrict__ src, const int* __restrict__ dst,
    const float* __restrict__ p, float* __restrict__ acc, int E)
{
    long gid = (long)blockIdx.x * 256 + threadIdx.x;
    int e = (int)(gid >> 5);
    int f = (int)(gid & 31);
    if (e < E) {
        int s = src[e];
        int d = dst[e];
        atomicAdd(&acc[(size_t)d * HID + f], p[(size_t)s * HID + f]);
    }
}

__global__ void deg_kernel(const int* __restrict__ dst, float* __restrict__ deg, int E)
{
    int e = blockIdx.x * 256 + threadIdx.x;
    if (e < E) atomicAdd(&deg[dst[e]], 1.0f);
}

__global__ void inv_kernel(const float* __restrict__ deg, float* __restrict__ inv, int N)
{
    int i = blockIdx.x * 256 + threadIdx.x;
    if (i < N) inv[i] = 1.0f / fmaxf(deg[i], 1.0f);
}

__global__ void zero_kernel(float* __restrict__ p, long n)
{
    long i = (long)blockIdx.x * 256 + threadIdx.x;
    if (i < n) p[i] = 0.0f;
}

// out = relu(acc * invdeg[n] + q + b[f])
__global__ void combine_kernel(const float* __restrict__ acc,
                               const float* __restrict__ invd,
                               const float* __restrict__ q,
                               const float* __restrict__ b,
                               float* __restrict__ out, int N)
{
    long gid = (long)blockIdx.x * 256 + threadIdx.x;
    if (gid < (long)N * HID) {
        int n = (int)(gid >> 5);
        int f = (int)(gid & 31);
        float v = acc[gid] * invd[n] + q[gid] + b[f];
        out[gid] = fmaxf(v, 0.0f);
    }
}

// Final head: [N,32] @ [32,2] + b — trivially VALU-bound.
__global__ void cls_kernel(const float* __restrict__ x,
                           const float* __restrict__ wout,
                           const float* __restrict__ bout,
                           float* __restrict__ out, int N)
{
    int n = blockIdx.x * 256 + threadIdx.x;
    if (n >= N) return;
    const float* xr = x + (size_t)n * HID;
    float o0 = bout[0], o1 = bout[1];
    for (int k = 0; k < HID; ++k) {
        float v = xr[k];
        o0 += v * wout[k * 2 + 0];
        o1 += v * wout[k * 2 + 1];
    }
    out[(size_t)n * 2 + 0] = o0;
    out[(size_t)n * 2 + 1] = o1;
}

// ---------------------------------------------------------------------------
// Host orchestration
// ---------------------------------------------------------------------------
static inline void run_gemm(const float* X, const float* W, float* Y,
                            int N, int K, hipStream_t s)
{
    gemm_x32_wmma<<<N / 128, 256, 0, s>>>(X, W, Y, N, K);
}

// One SAGE pass with linearity trick:
//   out = relu( segmean(Xsrc @ Wl)[edges] + Xdst @ Wr + b )
static void run_sage(const float* Xsrc, int Nsrc, int Ksrc,
                     const float* Xdst, int Ndst, int Kdst,
                     const float* Wl, const float* Wr, const float* b,
                     const int* esrc, const int* edst,
                     const float* invdeg_dst,
                     float* p, float* q, float* acc, float* out,
                     hipStream_t s)
{
    run_gemm(Xsrc, Wl, p, Nsrc, Ksrc, s);
    run_gemm(Xdst, Wr, q, Ndst, Kdst, s);
    zero_kernel<<<(int)(((long)Ndst * HID + 255) / 256), 256, 0, s>>>(acc, (long)Ndst * HID);
    scatter_kernel<<<(int)(((long)NEDGE * 32 + 255) / 256), 256, 0, s>>>(esrc, edst, p, acc, NEDGE);
    combine_kernel<<<(int)(((long)Ndst * HID + 255) / 256), 256, 0, s>>>(acc, invdeg_dst, q, b, out, Ndst);
}

extern "C" void kernel_launch(void* const* d_in, const int* in_sizes, int n_in,
                              void* d_out, int out_size, void* d_ws, size_t ws_size,
                              hipStream_t stream)
{
    const float* x_addr = (const float*)d_in[0];
    const float* x_tx   = (const float*)d_in[1];
    const int*   ei_at  = (const int*)d_in[2];   // [2,E]: src=addr, dst=tx
    const int*   ei_ta  = (const int*)d_in[3];   // [2,E]: src=tx,  dst=addr
    const float* w1_tx_l = (const float*)d_in[4];
    const float* b1_tx   = (const float*)d_in[5];
    const float* w1_tx_r = (const float*)d_in[6];
    const float* w1_ad_l = (const float*)d_in[7];
    const float* b1_ad   = (const float*)d_in[8];
    const float* w1_ad_r = (const float*)d_in[9];
    const float* w2_tx_l = (const float*)d_in[10];
    const float* b2_tx   = (const float*)d_in[11];
    const float* w2_tx_r = (const float*)d_in[12];
    const float* w2_ad_l = (const float*)d_in[13];
    const float* b2_ad   = (const float*)d_in[14];
    const float* w2_ad_r = (const float*)d_in[15];
    // w3_tx_* (16..18) unused by reference (layer 3 only updates addr nodes)
    const float* w3_ad_l = (const float*)d_in[19];
    const float* b3_ad   = (const float*)d_in[20];
    const float* w3_ad_r = (const float*)d_in[21];
    const float* w_out   = (const float*)d_in[22];
    const float* b_out   = (const float*)d_in[23];
    float* out = (float*)d_out;

    const int* at_src = ei_at;           const int* at_dst = ei_at + NEDGE;
    const int* ta_src = ei_ta;           const int* ta_dst = ei_ta + NEDGE;

    // Workspace layout (floats)
    float* ws = (float*)d_ws;
    const size_t FN = (size_t)N_ADDR * HID;     // 12.8M floats per node table
    float* invdeg_tx = ws;                      // N_TX
    float* invdeg_ad = invdeg_tx + N_TX;        // N_ADDR
    float* tx  = invdeg_ad + N_ADDR;            // FN
    float* ad  = tx  + FN;
    float* tx2 = ad  + FN;
    float* ad2 = tx2 + FN;
    float* ad3 = ad2 + FN;
    float* p   = ad3 + FN;                      // scratch: projected src feats
    float* q   = p   + FN;                      // scratch: dst transform
    float* acc = q   + FN;                      // scratch: edge accumulator

    const int EB = (NEDGE + 255) / 256;
    const int NB = (N_ADDR + 255) / 256;

    // Degrees (reuse acc as raw-count buffer)
    zero_kernel<<<NB, 256, 0, stream>>>(acc, (long)N_TX);
    deg_kernel<<<EB, 256, 0, stream>>>(at_dst, acc, NEDGE);
    inv_kernel<<<NB, 256, 0, stream>>>(acc, invdeg_tx, N_TX);
    zero_kernel<<<NB, 256, 0, stream>>>(acc, (long)N_ADDR);
    deg_kernel<<<EB, 256, 0, stream>>>(ta_dst, acc, NEDGE);
    inv_kernel<<<NB, 256, 0, stream>>>(acc, invdeg_ad, N_ADDR);

    // Layer 1 (inputs F_IN=64)
    run_sage(x_addr, N_ADDR, F_IN, x_tx, N_TX, F_IN,
             w1_tx_l, w1_tx_r, b1_tx, at_src, at_dst, invdeg_tx,
             p, q, acc, tx, stream);
    run_sage(x_tx, N_TX, F_IN, x_addr, N_ADDR, F_IN,
             w1_ad_l, w1_ad_r, b1_ad, ta_src, ta_dst, invdeg_ad,
             p, q, acc, ad, stream);
    // Layer 2 (HID=32)
    run_sage(ad, N_ADDR, HID, tx, N_TX, HID,
             w2_tx_l, w2_tx_r, b2_tx, at_src, at_dst, invdeg_tx,
             p, q, acc, tx2, stream);
    run_sage(tx, N_TX, HID, ad, N_ADDR, HID,
             w2_ad_l, w2_ad_r, b2_ad, ta_src, ta_dst, invdeg_ad,
             p, q, acc, ad2, stream);
    // Layer 3 (addr only)
    run_sage(tx2, N_TX, HID, ad2, N_ADDR, HID,
             w3_ad_l, w3_ad_r, b3_ad, ta_src, ta_dst, invdeg_ad,
             p, q, acc, ad3, stream);

    // Classifier head
    cls_kernel<<<NB, 256, 0, stream>>>(ad3, w_out, b_out, out, N_ADDR);
}